// CrossAttention_85890755985685
// MI455X (gfx1250) — compile-verified
//
#include <hip/hip_runtime.h>

// ============================================================================
// CrossAttention (collapsed form) for MI455X / gfx1250.
//
// Math: out = einsum('bij,bkv->biv', softmax(qk), v) factorizes into
//   (rowsum softmax) * (colsum v)  -> the positive per-row scalar cancels
// under the L2 normalize, so  output = x + Sv/max(||Sv||,1e-12)  with
//   Sv[b,d] = (sum_k y_res[b,k,:]) . Wv[d,:] + SEQ*bv[d].
// Bandwidth-bound: ~100 MB traffic -> ~4.5us @ 23.3 TB/s. The remaining GEMM
// (ysum[4x1024] x Wv^T[1024x1024]) runs on V_WMMA_F32_16X16X4_F32.
// ============================================================================

#define DIMN   1024
#define BATCH  4
#define SEQ    2048
#define KSPLIT 16
#define ROWS_PER_SPLIT (SEQ / KSPLIT)   // 128

typedef __attribute__((ext_vector_type(2))) float v2f;
typedef __attribute__((ext_vector_type(8))) float v8f;

// Workspace layout (float offsets)
#define WS_PART  0                               // KSPLIT*BATCH*DIMN
#define WS_YSUM  (KSPLIT * BATCH * DIMN)         // 16*DIMN (rows 4..15 zero)
#define WS_SV    (WS_YSUM + 16 * DIMN)           // BATCH*DIMN
#define WS_SCALE (WS_SV + BATCH * DIMN)          // BATCH

// --- 1) split-K column sum of y_res: part[z][b][e] = sum over 128 rows -----
__global__ void colsum_partial(const float* __restrict__ y, float* __restrict__ part) {
    const int e = blockIdx.x * blockDim.x + threadIdx.x;   // 0..1023
    const int b = blockIdx.y;
    const int z = blockIdx.z;
    const float* p = y + ((size_t)b * SEQ + (size_t)z * ROWS_PER_SPLIT) * DIMN + e;
    float s = 0.0f;
#pragma unroll 8
    for (int k = 0; k < ROWS_PER_SPLIT; ++k) s += p[(size_t)k * DIMN];
    part[(z * BATCH + b) * DIMN + e] = s;
}

// --- 2) reduce partials into zero-padded 16-row A matrix -------------------
__global__ void reduce_partials(const float* __restrict__ part, float* __restrict__ ysum) {
    const int i = blockIdx.x * blockDim.x + threadIdx.x;   // 0 .. 16*DIMN-1
    const int m = i >> 10;          // padded row
    const int e = i & (DIMN - 1);
    float s = 0.0f;
    if (m < BATCH) {
#pragma unroll
        for (int z = 0; z < KSPLIT; ++z) s += part[(z * BATCH + m) * DIMN + e];
    }
    ysum[i] = s;                    // rows 4..15 = 0 (keeps WMMA EXEC all-1s)
}

// --- 3) Sv = ysum_pad(16x1024) x Wv^T via V_WMMA_F32_16X16X4_F32 -----------
// A 16x4 layout (ISA 7.12.2): lane<16 -> M=lane, VGPR0/1 = K0/K1;
//                             lane>=16 -> M=lane-16, VGPR0/1 = K2/K3.
// B 4x16 mirrored: lane%16 = N, (lane/16) selects K-pair. B[k][n] = Wv[n][k].
__global__ void __launch_bounds__(32) sv_wmma(const float* __restrict__ ysum, // [16][1024]
                                              const float* __restrict__ Wv,   // [1024][1024]
                                              const float* __restrict__ bv,   // [1024]
                                              float* __restrict__ Sv) {       // [4][1024]
    const int n0   = blockIdx.x * 16;      // output-column tile
    const int lane = threadIdx.x;
    const int half = lane >> 4;            // 0: K={0,1}, 1: K={2,3}
    const int l16  = lane & 15;

    const float* arow = ysum + (size_t)l16 * DIMN + 2 * half;          // row M=l16
    const float* brow = Wv + (size_t)(n0 + l16) * DIMN + 2 * half;     // row N=n0+l16

    v8f acc = {};
    for (int k = 0; k < DIMN; k += 4) {
        v2f a = *(const v2f*)(arow + k);
        v2f b = *(const v2f*)(brow + k);
        acc = __builtin_amdgcn_wmma_f32_16x16x4_f32(
            /*neg_a=*/false, a, /*neg_b=*/false, b,
            /*c_mod=*/(short)0, acc, /*reuse_a=*/false, /*reuse_b=*/false);
    }
    // D layout: VGPR j, lanes 0-15 hold (M=j, N=lane). Rows 0..3 are batches.
    if (half == 0) {
        const float bb = bv[n0 + l16] * (float)SEQ;
#pragma unroll
        for (int j = 0; j < BATCH; ++j)
            Sv[j * DIMN + n0 + l16] = acc[j] + bb;
    }
}

// --- 4) per-batch inverse L2 norm ------------------------------------------
__global__ void norm_scale(const float* __restrict__ Sv, float* __restrict__ scale) {
    __shared__ float red[256];
    const int b = blockIdx.x;
    const int t = threadIdx.x;
    float s = 0.0f;
    for (int d = t; d < DIMN; d += 256) { float v = Sv[b * DIMN + d]; s += v * v; }
    red[t] = s;
    __syncthreads();
    for (int off = 128; off > 0; off >>= 1) {
        if (t < off) red[t] += red[t + off];
        __syncthreads();
    }
    if (t == 0) scale[b] = 1.0f / fmaxf(sqrtf(red[0]), 1e-12f);
}

// --- 5) out = x + Sv[b,:] * scale[b]   (float4-vectorized, BW-bound) -------
__global__ void final_add(const float* __restrict__ x, const float* __restrict__ Sv,
                          const float* __restrict__ scale, float* __restrict__ out) {
    const size_t i4   = (size_t)blockIdx.x * blockDim.x + threadIdx.x;
    const size_t base = i4 * 4;                 // flat float index
    const int d = (int)(base & (DIMN - 1));
    const int b = (int)(base >> 21);            // / (SEQ*DIMN) = / 2^21
    const float sc = scale[b];
    const float4 xv = *(const float4*)(x + base);
    const float4 sv = *(const float4*)(Sv + b * DIMN + d);
    float4 o;
    o.x = xv.x + sv.x * sc;
    o.y = xv.y + sv.y * sc;
    o.z = xv.z + sv.z * sc;
    o.w = xv.w + sv.w * sc;
    *(float4*)(out + base) = o;
}

extern "C" void kernel_launch(void* const* d_in, const int* in_sizes, int n_in,
                              void* d_out, int out_size, void* d_ws, size_t ws_size,
                              hipStream_t stream) {
    const float* x     = (const float*)d_in[0];
    // d_in[1]=x_res, d_in[3..6]=Wq,bq,Wk,bk: algebraically eliminated
    // (softmax row-sums are positive scalars cancelled by the L2 normalize).
    const float* y_res = (const float*)d_in[2];
    const float* Wv    = (const float*)d_in[7];
    const float* bv    = (const float*)d_in[8];
    float* out = (float*)d_out;

    float* ws    = (float*)d_ws;
    float* part  = ws + WS_PART;
    float* ysum  = ws + WS_YSUM;
    float* Sv    = ws + WS_SV;
    float* scale = ws + WS_SCALE;

    (void)in_sizes; (void)n_in; (void)out_size; (void)ws_size;

    colsum_partial<<<dim3(DIMN / 256, BATCH, KSPLIT), 256, 0, stream>>>(y_res, part);
    reduce_partials<<<dim3(16 * DIMN / 256), 256, 0, stream>>>(part, ysum);
    sv_wmma<<<dim3(DIMN / 16), 32, 0, stream>>>(ysum, Wv, bv, Sv);
    norm_scale<<<dim3(BATCH), 256, 0, stream>>>(Sv, scale);
    final_add<<<dim3((BATCH * SEQ * DIMN / 4) / 256), 256, 0, stream>>>(x, Sv, scale, out);
}